// LinkPredictionModel_79963701117032
// MI455X (gfx1250) — compile-verified
//
#include <hip/hip_runtime.h>
#include <hip/hip_bf16.h>
#include <cstddef>

#define NN 100000

typedef float v2f __attribute__((ext_vector_type(2)));
typedef float v8f __attribute__((ext_vector_type(8)));

// ---------------------------------------------------------------- degree ----
__global__ void k_zero(float* __restrict__ p, int n) {
  int i = blockIdx.x * blockDim.x + threadIdx.x;
  if (i < n) p[i] = 0.0f;
}

__global__ void k_deg(const int* __restrict__ dst, int E, float* __restrict__ deg) {
  int e = blockIdx.x * blockDim.x + threadIdx.x;
  if (e < E) atomicAdd(&deg[dst[e]], 1.0f);
}

__global__ void k_dinv(float* __restrict__ d, int n) {
  int i = blockIdx.x * blockDim.x + threadIdx.x;
  if (i < n) d[i] = rsqrtf(d[i] + 1.0f);
}

// ------------------------------------------------- fp32 WMMA GEMM (M,K)@(K,N)
// One wave computes a 16x16 output tile; B tile staged transposed in LDS,
// zero-padded to 16 columns. A fragment: 16x4 f32 layout — lanes 0-15 hold
// M=0..15 with K pair {0,1}; lanes 16-31 hold K pair {2,3}. B mirrors with N.
template<int K, int NCOL, int WPB>
__global__ __launch_bounds__(WPB * 32)
void k_gemm_wmma(const float* __restrict__ A, const float* __restrict__ B,
                 float* __restrict__ C, int Mrows) {
  constexpr int KP = K + 4;                 // pad -> conflict-free ds_load_b64
  __shared__ float Bt[16 * KP];
  const int nbase = blockIdx.y * 16;
  for (int idx = threadIdx.x; idx < 16 * K; idx += WPB * 32) {
    const int n  = idx / K;
    const int k  = idx - n * K;
    const int ng = nbase + n;
    Bt[n * KP + k] = (ng < NCOL) ? B[(size_t)k * NCOL + ng] : 0.0f;
  }
  __syncthreads();

  const int lane  = threadIdx.x & 31;
  const int wave  = threadIdx.x >> 5;
  const int mtile = blockIdx.x * WPB + wave;
  const int row0  = mtile * 16;
  if (row0 >= Mrows) return;                // wave-uniform: EXEC stays all-1s

  const int l15  = lane & 15;
  const int koff = (lane >> 4) << 1;        // 0 or 2
  const float* __restrict__ arow = A + (size_t)(row0 + l15) * K + koff;
  const float* brow = &Bt[l15 * KP + koff];

  v8f acc = {0.f, 0.f, 0.f, 0.f, 0.f, 0.f, 0.f, 0.f};
#pragma unroll 4
  for (int kb = 0; kb < K; kb += 4) {
    v2f a = *(const v2f*)(arow + kb);
    v2f b = *(const v2f*)(brow + kb);
    acc = __builtin_amdgcn_wmma_f32_16x16x4_f32(
        /*neg_a=*/false, a, /*neg_b=*/false, b,
        /*c_mod=*/(short)0, acc, /*reuse_a=*/false, /*reuse_b=*/false);
  }

  const int mo  = (lane >> 4) << 3;         // 0 or 8
  const int col = nbase + l15;
  if (col < NCOL) {
    float* __restrict__ crow = C + (size_t)(row0 + mo) * NCOL + col;
#pragma unroll
    for (int r = 0; r < 8; ++r) crow[(size_t)r * NCOL] = acc[r];
  }
}

// -------------------------------------------- self term + bias (init h) ----
template<int F>
__global__ void k_self_bias(const float* __restrict__ xw, const float* __restrict__ dinv,
                            const float* __restrict__ bias, float* __restrict__ h, int n) {
  int i = blockIdx.x * blockDim.x + threadIdx.x;
  if (i < n * F) {
    int v = i / F;
    int f = i - v * F;
    float d = dinv[v];
    h[i] = xw[i] * (d * d) + bias[f];
  }
}

// ------------------------------------------------- edge scatter (atomic) ----
template<int F>
__global__ void k_edge(const int* __restrict__ src, const int* __restrict__ dst, int E,
                       const float* __restrict__ xw, const float* __restrict__ dinv,
                       float* __restrict__ h) {
  int e = blockIdx.x * blockDim.x + threadIdx.x;
  if (e >= E) return;
  const int s = src[e];
  const int d = dst[e];
  const float nrm = dinv[s] * dinv[d];
  const float4* __restrict__ xs = (const float4*)(xw + (size_t)s * F);
  float* __restrict__ hd = h + (size_t)d * F;
#pragma unroll
  for (int c = 0; c < F / 4; ++c) {
    float4 v = xs[c];
    atomicAdd(hd + 4 * c + 0, v.x * nrm);
    atomicAdd(hd + 4 * c + 1, v.y * nrm);
    atomicAdd(hd + 4 * c + 2, v.z * nrm);
    atomicAdd(hd + 4 * c + 3, v.w * nrm);
  }
}

__global__ void k_relu(float* __restrict__ h, int n) {
  int i = blockIdx.x * blockDim.x + threadIdx.x;
  if (i < n) h[i] = fmaxf(h[i], 0.0f);
}

// ----------------------------------------------------- final prediction ----
__global__ void k_predict(const int* __restrict__ smp, int S,
                          const float* __restrict__ h3,
                          const float* __restrict__ Wfc, const float* __restrict__ bfc,
                          float* __restrict__ out) {
  int si = blockIdx.x * blockDim.x + threadIdx.x;
  if (si >= S) return;
  const int i = smp[2 * si];
  const int j = smp[2 * si + 1];
  const float4* __restrict__ hi = (const float4*)(h3 + (size_t)i * 32);
  const float4* __restrict__ hj = (const float4*)(h3 + (size_t)j * 32);
  const float4* __restrict__ w  = (const float4*)Wfc;
  float z = bfc[0];
#pragma unroll
  for (int c = 0; c < 8; ++c) {
    float4 a = hi[c], ww = w[c];
    z += a.x * ww.x + a.y * ww.y + a.z * ww.z + a.w * ww.w;
  }
#pragma unroll
  for (int c = 0; c < 8; ++c) {
    float4 a = hj[c], ww = w[8 + c];
    z += a.x * ww.x + a.y * ww.y + a.z * ww.z + a.w * ww.w;
  }
  out[si] = 1.0f / (1.0f + expf(-z));
}

// ---------------------------------------------------------------------------
extern "C" void kernel_launch(void* const* d_in, const int* in_sizes, int n_in,
                              void* d_out, int out_size, void* d_ws, size_t ws_size,
                              hipStream_t stream) {
  const float* x   = (const float*)d_in[0];
  const int* eidx  = (const int*)d_in[1];
  const int* smp   = (const int*)d_in[2];
  const float* W1  = (const float*)d_in[3];
  const float* b1  = (const float*)d_in[4];
  const float* W2  = (const float*)d_in[5];
  const float* b2  = (const float*)d_in[6];
  const float* W3  = (const float*)d_in[7];
  const float* b3  = (const float*)d_in[8];
  const float* Wfc = (const float*)d_in[9];
  const float* bfc = (const float*)d_in[10];
  float* out = (float*)d_out;

  const int E = in_sizes[1] / 2;
  const int S = in_sizes[2] / 2;
  (void)n_in; (void)out_size; (void)ws_size;

  const int* src = eidx;
  const int* dst = eidx + E;

  // workspace layout (floats): dinv | slotA(3.2M) | slotB(3.2M) | slotC(1.6M)
  float* ws    = (float*)d_ws;
  float* dinv  = ws;
  float* slotA = ws + 102400;
  float* slotB = slotA + 3200000;
  float* slotC = slotB + 3200000;

  const int TB = 256;
  const int gN   = (NN + TB - 1) / TB;
  const int gE   = (E + TB - 1) / TB;
  const int MT   = (NN + 15) / 16;          // 6250 m-tiles
  const int WPB  = 4;
  const int gMT  = (MT + WPB - 1) / WPB;

  // degree -> dinv
  k_zero<<<gN, TB, 0, stream>>>(dinv, NN);
  k_deg <<<gE, TB, 0, stream>>>(dst, E, dinv);
  k_dinv<<<gN, TB, 0, stream>>>(dinv, NN);

  // ---- layer 1: xw1 = x @ W1 (K=512, N=8) -> slotA ; h1 -> slotB
  k_gemm_wmma<512, 8, 4><<<dim3(gMT, 1), WPB * 32, 0, stream>>>(x, W1, slotA, NN);
  k_self_bias<8><<<(NN * 8 + TB - 1) / TB, TB, 0, stream>>>(slotA, dinv, b1, slotB, NN);
  k_edge<8><<<gE, TB, 0, stream>>>(src, dst, E, slotA, dinv, slotB);
  k_relu<<<(NN * 8 + TB - 1) / TB, TB, 0, stream>>>(slotB, NN * 8);

  // ---- layer 2: xw2 = h1 @ W2 (K=8, N=16) -> slotA ; h2 -> slotC
  k_gemm_wmma<8, 16, 4><<<dim3(gMT, 1), WPB * 32, 0, stream>>>(slotB, W2, slotA, NN);
  k_self_bias<16><<<(NN * 16 + TB - 1) / TB, TB, 0, stream>>>(slotA, dinv, b2, slotC, NN);
  k_edge<16><<<gE, TB, 0, stream>>>(src, dst, E, slotA, dinv, slotC);
  k_relu<<<(NN * 16 + TB - 1) / TB, TB, 0, stream>>>(slotC, NN * 16);

  // ---- layer 3: xw3 = h2 @ W3 (K=16, N=32) -> slotB ; h3 -> slotA
  k_gemm_wmma<16, 32, 4><<<dim3(gMT, 2), WPB * 32, 0, stream>>>(slotC, W3, slotB, NN);
  k_self_bias<32><<<(NN * 32 + TB - 1) / TB, TB, 0, stream>>>(slotB, dinv, b3, slotA, NN);
  k_edge<32><<<gE, TB, 0, stream>>>(src, dst, E, slotB, dinv, slotA);

  // ---- link prediction
  k_predict<<<(S + TB - 1) / TB, TB, 0, stream>>>(smp, S, slotA, Wfc, bfc, out);
}